// LeibnizCoupling_17763984736922
// MI455X (gfx1250) — compile-verified
//
#include <hip/hip_runtime.h>

typedef __attribute__((ext_vector_type(2))) float v2f;
typedef __attribute__((ext_vector_type(4))) float v4f;
typedef __attribute__((ext_vector_type(8))) float v8f;

#define FDIM   32
#define NPATH  12
#define TILE   16
#define WAVES  4

// Scheduling cut: IR-level memory clobber (blocks load CSE/motion) plus a
// machine-scheduler barrier (blocks WMMA/VALU motion). Bounds register live
// ranges so allocation stays well under 256 VGPRs (no MSB mode, no spills).
#if __has_builtin(__builtin_amdgcn_sched_barrier)
#define CUT()                        \
  do {                               \
    asm volatile("" ::: "memory");   \
    __builtin_amdgcn_sched_barrier(0); \
  } while (0)
#else
#define CUT() asm volatile("" ::: "memory")
#endif

__device__ __forceinline__ v8f wmma4(v2f a, v2f b, v8f c) {
  // D = A(16x4 f32) x B(4x16 f32) + C(16x16 f32)
  return __builtin_amdgcn_wmma_f32_16x16x4_f32(false, a, false, b, (short)0, c,
                                               false, false);
}

__global__ __launch_bounds__(WAVES * 32)
void leibniz_coupling_kernel(const float* __restrict__ h0,
                             const float* __restrict__ h1,
                             const float* __restrict__ h2,
                             const float* __restrict__ g0,
                             const float* __restrict__ g1,
                             const float* __restrict__ g2,
                             const float* __restrict__ W,
                             const long long* __restrict__ src,
                             float* __restrict__ out0,
                             float* __restrict__ out1,
                             float* __restrict__ out2) {
  __shared__ float sW[NPATH * FDIM * FDIM];  // 48 KB, all 12 path weights
  {
    const v4f* wv = (const v4f*)W;
    v4f* sv = (v4f*)sW;
    for (int t = threadIdx.x; t < NPATH * FDIM * FDIM / 4; t += blockDim.x)
      sv[t] = wv[t];
  }
  __syncthreads();

  const int lane = threadIdx.x & 31;
  const int wave = threadIdx.x >> 5;
  const int lo16 = lane & 15;   // A row / feature-column index
  const int hi   = lane >> 4;   // K-pair / M-half selector
  const int e0   = (blockIdx.x * WAVES + wave) * TILE;

  // gather node row for this lane's A-matrix row (lanes 16..31 mirror 0..15)
  const long long srow = src[e0 + lo16];
  const float* __restrict__ ah0 = h0 + (size_t)srow * FDIM;
  const float* __restrict__ ah1 = h1 + (size_t)srow * 3 * FDIM;
  const float* __restrict__ ah2 = h2 + (size_t)srow * 9 * FDIM;

  const v8f vzero = {0.f, 0.f, 0.f, 0.f, 0.f, 0.f, 0.f, 0.f};
  const float INV_SQRT_F = 0.17677669529663687f;  // 32^-0.5

  v2f Bf[8];  // B fragments for current (path, N-half): [kstep]
  auto load_B = [&](int p, int half) {
#pragma unroll
    for (int s = 0; s < 8; ++s)
      Bf[s] =
          *(const v2f*)&sW[(p * FDIM + half * 16 + lo16) * FDIM + 4 * s + 2 * hi];
  };

  // T = (h-rows) x W[p]^T for one geometric component, one N-half
  auto gemm1 = [&](const float* arow) -> v8f {
    v8f d = vzero;
#pragma unroll
    for (int s = 0; s < 8; ++s) {
      v2f a = *(const v2f*)(arow + 4 * s + 2 * hi);
      d = wmma4(a, Bf[s], d);
    }
    return d;
  };

  // g[e0+i+8*hi][comp][lo16+16*half] for i=0..7  (2 x 64B coalesced per load)
  auto load_g = [&](const float* gbase, int ncomp, int comp, int half) -> v8f {
    const float* p = gbase + ((size_t)(e0 + 8 * hi) * ncomp + comp) * FDIM +
                     lo16 + 16 * half;
    const size_t stride = (size_t)ncomp * FDIM;
    v8f r;
#pragma unroll
    for (int i = 0; i < 8; ++i) r[i] = p[(size_t)i * stride];
    return r;
  };

  auto store_out = [&](float* obase, int ncomp, int comp, int half, v8f v) {
    float* p = obase + ((size_t)(e0 + 8 * hi) * ncomp + comp) * FDIM +
               lo16 + 16 * half;
    const size_t stride = (size_t)ncomp * FDIM;
#pragma unroll
    for (int i = 0; i < 8; ++i)
      __builtin_nontemporal_store(v[i] * INV_SQRT_F, p + (size_t)i * stride);
  };

  // Runtime loop over the two 16-feature output halves: halves accumulator
  // pressure AND static code size (I$), same total WMMA work.
#pragma unroll 1
  for (int half = 0; half < 2; ++half) {
    // -------- lo = 0 : paths 0 (prod 0,0), 5 (dot 1,1), 10 (ddot 2,2)
    {
      v8f o = vzero;
      load_B(0, half);
      o += gemm1(ah0) * load_g(g0, 1, 0, half);
      CUT();
      load_B(5, half);
#pragma unroll
      for (int i = 0; i < 3; ++i) {
        o += gemm1(ah1 + i * FDIM) * load_g(g1, 3, i, half);
        CUT();
      }
      load_B(10, half);
#pragma unroll
      for (int c = 0; c < 9; ++c) {
        o += gemm1(ah2 + c * FDIM) * load_g(g2, 9, c, half);
        CUT();
      }
      store_out(out0, 1, 0, half, o);
      CUT();
    }

    // -------- lo = 1 : paths 1, 3, 6 (cross), 8 (mat_vec), 9 (vec_mat)
    {
      v8f o[3];
#pragma unroll
      for (int i = 0; i < 3; ++i) o[i] = vzero;

      load_B(1, half);  // prod(0,1)
      {
        v8f t = gemm1(ah0);
#pragma unroll
        for (int i = 0; i < 3; ++i) o[i] += t * load_g(g1, 3, i, half);
      }
      CUT();

      load_B(3, half);  // prod(1,0)
      {
        v8f q = load_g(g0, 1, 0, half);
#pragma unroll
        for (int i = 0; i < 3; ++i) {
          o[i] += gemm1(ah1 + i * FDIM) * q;
          CUT();
        }
      }

      load_B(6, half);  // cross(1,1): res[i] = a[i+1]b[i+2] - a[i+2]b[i+1]
#pragma unroll
      for (int j = 0; j < 3; ++j) {
        v8f t = gemm1(ah1 + j * FDIM);
        const int jp = (j + 1) % 3, jpp = (j + 2) % 3;
        o[jpp] += t * load_g(g1, 3, jp, half);
        o[jp] -= t * load_g(g1, 3, jpp, half);
        CUT();
      }

      load_B(8, half);  // mat_vec(2,1): res[i] = sum_j T[i][j] g1[j]
#pragma unroll
      for (int i = 0; i < 3; ++i)
#pragma unroll
        for (int j = 0; j < 3; ++j) {
          o[i] += gemm1(ah2 + (3 * i + j) * FDIM) * load_g(g1, 3, j, half);
          CUT();
        }

      load_B(9, half);  // vec_mat(1,2): res[j] = sum_i T[i] g2[i][j]
#pragma unroll
      for (int i = 0; i < 3; ++i) {
        v8f t = gemm1(ah1 + i * FDIM);
#pragma unroll
        for (int j = 0; j < 3; ++j)
          o[j] += t * load_g(g2, 9, 3 * i + j, half);
        CUT();
      }

#pragma unroll
      for (int i = 0; i < 3; ++i) store_out(out1, 3, i, half, o[i]);
      CUT();
    }

    // -------- lo = 2 : paths 2, 4, 7 (outer), 11 (mat_mul_sym)
    {
      v8f o[9];
#pragma unroll
      for (int c = 0; c < 9; ++c) o[c] = vzero;
      v8f tr = vzero;  // shared trace accumulator (p7 + p11)

      load_B(2, half);  // prod(0,2)
      {
        v8f t = gemm1(ah0);
#pragma unroll
        for (int c = 0; c < 9; ++c) {
          o[c] += t * load_g(g2, 9, c, half);
          if (c % 3 == 2) CUT();  // t (8 regs) stays live; g window bounded
        }
      }
      CUT();

      load_B(4, half);  // prod(2,0)
      {
        v8f q = load_g(g0, 1, 0, half);
#pragma unroll
        for (int c = 0; c < 9; ++c) {
          o[c] += gemm1(ah2 + c * FDIM) * q;
          CUT();
        }
      }

      load_B(7, half);  // outer(1,1) - (1/3)tr*I
#pragma unroll
      for (int i = 0; i < 3; ++i) {
        v8f t = gemm1(ah1 + i * FDIM);
#pragma unroll
        for (int j = 0; j < 3; ++j) {
          v8f v = t * load_g(g1, 3, j, half);
          o[3 * i + j] += v;
          if (i == j) tr += v;
        }
        CUT();
      }

      load_B(11, half);  // mat_mul_sym(2,2): raw[a][j] = sum_k T[a][k] g2[k][j]
#pragma unroll
      for (int a = 0; a < 3; ++a)
#pragma unroll
        for (int k = 0; k < 3; ++k) {
          v8f t = gemm1(ah2 + (3 * a + k) * FDIM);
#pragma unroll
          for (int j = 0; j < 3; ++j) {
            v8f v = t * load_g(g2, 9, 3 * k + j, half);
            if (j == a) {  // diagonal: 0.5v + 0.5v = v; also feeds the trace
              o[3 * a + j] += v;
              tr += v;
            } else {
              o[3 * a + j] += 0.5f * v;
              o[3 * j + a] += 0.5f * v;
            }
          }
          CUT();
        }

      const float third = 1.0f / 3.0f;
#pragma unroll
      for (int d = 0; d < 3; ++d) o[4 * d] -= third * tr;
#pragma unroll
      for (int c = 0; c < 9; ++c) store_out(out2, 9, c, half, o[c]);
      CUT();
    }
  }
}

extern "C" void kernel_launch(void* const* d_in, const int* in_sizes, int n_in,
                              void* d_out, int out_size, void* d_ws, size_t ws_size,
                              hipStream_t stream) {
  const float* h0 = (const float*)d_in[0];
  const float* h1 = (const float*)d_in[1];
  const float* h2 = (const float*)d_in[2];
  const float* g0 = (const float*)d_in[3];
  const float* g1 = (const float*)d_in[4];
  const float* g2 = (const float*)d_in[5];
  const float* W  = (const float*)d_in[6];
  const long long* src = (const long long*)d_in[7];

  const int E = in_sizes[7];  // src has one entry per edge
  float* out0 = (float*)d_out;
  float* out1 = out0 + (size_t)E * FDIM;
  float* out2 = out1 + (size_t)E * 3 * FDIM;

  const int blocks = E / (TILE * WAVES);  // E = 131072 -> 2048 blocks, no tail
  leibniz_coupling_kernel<<<blocks, WAVES * 32, 0, stream>>>(
      h0, h1, h2, g0, g1, g2, W, src, out0, out1, out2);
}